// Rasterize_4037269259130
// MI455X (gfx1250) — compile-verified
//
#include <hip/hip_runtime.h>
#include <hip/hip_bf16.h>

typedef __attribute__((ext_vector_type(2))) float v2f;
typedef __attribute__((ext_vector_type(8))) float v8f;

#define IS 256
#define BB 4
#define FF 4096
#define TT 4
#define NEARZ 0.1f
#define FARZ 100.0f
#define TINYF 1e-12f
#define NT 4   // pixel tiles (of 16) per wave

__device__ __forceinline__ float pixc(int j) {
    return (2.0f * (float)j + 1.0f - (float)IS) / (float)IS;
}

// ---------------------------------------------------------------------------
// Pass 0: per-face edge-plane coefficients.
// Row of 16 floats per face:
// [a0,b0,c0, a1,b1,c1, a2,b2,c2, da,db,dc, pa,pb,pc, 0]
//  w_k  = a_k*x + b_k*y + c_k
//  det  = da*x + db*y + dc          (= w0+w1+w2)
//  dot  = pa*x + pb*y + pc          (= w0/z0 + w1/z1 + w2/z2)
// ---------------------------------------------------------------------------
__global__ __launch_bounds__(256) void prep_kernel(const float* __restrict__ faces,
                                                   float* __restrict__ prep) {
    int t = blockIdx.x * blockDim.x + threadIdx.x;
    if (t >= BB * FF) return;
    const float* f = faces + t * 9;
    float x0 = f[0], y0 = f[1], z0 = f[2];
    float x1 = f[3], y1 = f[4], z1 = f[5];
    float x2 = f[6], y2 = f[7], z2 = f[8];
    float a0 = y1 - y2, b0 = x2 - x1, c0 = x1 * y2 - x2 * y1;
    float a1 = y2 - y0, b1 = x0 - x2, c1 = x2 * y0 - x0 * y2;
    float a2 = y0 - y1, b2 = x1 - x0, c2 = x0 * y1 - x1 * y0;
    float r0 = 1.0f / z0, r1 = 1.0f / z1, r2 = 1.0f / z2;
    float* p = prep + t * 16;
    p[0]  = a0; p[1]  = b0; p[2]  = c0;
    p[3]  = a1; p[4]  = b1; p[5]  = c1;
    p[6]  = a2; p[7]  = b2; p[8]  = c2;
    p[9]  = a0 + a1 + a2; p[10] = b0 + b1 + b2; p[11] = c0 + c1 + c2;
    p[12] = a0 * r0 + a1 * r1 + a2 * r2;
    p[13] = b0 * r0 + b1 * r1 + b2 * r2;
    p[14] = c0 * r0 + c1 * r1 + c2 * r2;
    p[15] = 0.0f;
}

// ---------------------------------------------------------------------------
// Pass 1: depth test via WMMA.  D = A(16 faces x [a,b,c,0]) * B([x;y;1;0] x 16 px)
// A layout (32-bit 16x4): lanes 0-15 hold K=0,1 in VGPR0,1 ; lanes 16-31 hold K=2,3.
// B layout (4x16): VGPR0 = row0(x) lanes0-15 / row1(y) lanes16-31; VGPR1 = 1 / 0.
// D layout: lane L, vgpr r -> face M = r + 8*(L>=16), pixel N = L&15.
// ---------------------------------------------------------------------------
__global__ __launch_bounds__(256) void depth_kernel(const float* __restrict__ prep,
                                                    int* __restrict__ fidx) {
    const int lane = threadIdx.x & 31;
    const int wv   = threadIdx.x >> 5;
    const int b    = blockIdx.x >> 7;     // 128 blocks per batch
    const int blk  = blockIdx.x & 127;
    const int tile0 = (blk * 8 + wv) * NT; // tile = 16 consecutive x in one row
    const bool hi  = lane >= 16;
    const int  lx  = lane & 15;

    v2f Bv[NT];
    int tilepix[NT];
#pragma unroll
    for (int t = 0; t < NT; ++t) {
        int tl = tile0 + t;
        int y  = tl >> 4;
        int x0 = (tl & 15) << 4;
        float xp = pixc(x0 + lx);
        float yp = pixc(y);
        Bv[t].x = hi ? yp : xp;
        Bv[t].y = hi ? 0.0f : 1.0f;
        tilepix[t] = b * IS * IS + tl * 16;
    }

    float best[NT];
    int   bidx[NT];
#pragma unroll
    for (int t = 0; t < NT; ++t) { best[t] = FARZ; bidx[t] = -1; }

    const float* pb = prep + b * FF * 16;
    const int faceOfs = hi ? 8 : 0;

    for (int g = 0; g < FF / 16; ++g) {
        const int gbase = g * 16;
        const float4* row = (const float4*)(pb + (gbase + lx) * 16);
        float4 c0 = row[0], c1 = row[1], c2 = row[2], c3 = row[3];
        if (g + 1 < FF / 16)
            __builtin_prefetch(pb + (gbase + 16 + lx) * 16, 0, 3);

        v2f A0, A1, A2, A3, A4;
        A0.x = hi ? c0.z : c0.x;  A0.y = hi ? 0.0f : c0.y;   // w0: a,b | c,0
        A1.x = hi ? c1.y : c0.w;  A1.y = hi ? 0.0f : c1.x;   // w1
        A2.x = hi ? c2.x : c1.z;  A2.y = hi ? 0.0f : c1.w;   // w2
        A3.x = hi ? c2.w : c2.y;  A3.y = hi ? 0.0f : c2.z;   // det
        A4.x = hi ? c3.z : c3.x;  A4.y = hi ? 0.0f : c3.y;   // dot (1/z weighted)

#pragma unroll
        for (int t = 0; t < NT; ++t) {
            v8f z = {};
            v8f d0 = __builtin_amdgcn_wmma_f32_16x16x4_f32(false, A0, false, Bv[t], (short)0, z, false, false);
            v8f d1 = __builtin_amdgcn_wmma_f32_16x16x4_f32(false, A1, false, Bv[t], (short)0, z, false, false);
            v8f d2 = __builtin_amdgcn_wmma_f32_16x16x4_f32(false, A2, false, Bv[t], (short)0, z, false, false);
            v8f d3 = __builtin_amdgcn_wmma_f32_16x16x4_f32(false, A3, false, Bv[t], (short)0, z, false, false);
            v8f d4 = __builtin_amdgcn_wmma_f32_16x16x4_f32(false, A4, false, Bv[t], (short)0, z, false, false);
#pragma unroll
            for (int r = 0; r < 8; ++r) {
                float w0 = d0[r], w1 = d1[r], w2 = d2[r];
                float det = d3[r], dot = d4[r];
                // branchless inside test: all weights strictly same (nonzero) sign
                float wmin = fminf(fminf(w0, w1), w2);
                float wmax = fmaxf(fmaxf(w0, w1), w2);
                // fast reciprocal: selection-only depth (shade pass recomputes exactly)
                float zp = det * __builtin_amdgcn_rcpf(dot);
                bool valid = (__builtin_fabsf(det) > TINYF)
                           & ((wmin > 0.0f) | (wmax < 0.0f))
                           & (zp > NEARZ) & (zp < FARZ);
                float cand = valid ? zp : 3.0e38f;
                int fid = gbase + faceOfs + r;
                bool tk = cand < best[t];          // strict <  => first-min wins
                best[t] = tk ? cand : best[t];
                bidx[t] = tk ? fid  : bidx[t];
            }
        }
    }

#pragma unroll
    for (int t = 0; t < NT; ++t) {
        float oz = __shfl_xor(best[t], 16, 32);
        int   oi = __shfl_xor(bidx[t], 16, 32);
        bool take = (oz < best[t]) | ((oz == best[t]) & (oi < bidx[t]));
        int i = take ? oi : bidx[t];
        if (!hi) fidx[tilepix[t] + lx] = i;
    }
}

// ---------------------------------------------------------------------------
// Pass 2: shading (faithful reference math incl. clamps), trilinear texture.
// ---------------------------------------------------------------------------
__global__ __launch_bounds__(256) void shade_kernel(const float* __restrict__ faces,
                                                    const float* __restrict__ tex,
                                                    const int* __restrict__ fidx,
                                                    float* __restrict__ out) {
    int p = blockIdx.x * blockDim.x + threadIdx.x;
    if (p >= BB * IS * IS) return;
    int b = p >> 16;
    int rem = p & 65535;
    int y = rem >> 8, x = rem & 255;
    float xp = pixc(x), yp = pixc(y);
    float* o = out + p * 5;

    int f = fidx[p];
    if (f < 0) {
        o[0] = 0.0f; o[1] = 0.0f; o[2] = 0.0f; o[3] = 0.0f; o[4] = FARZ;
        return;
    }
    const float* fv = faces + (b * FF + f) * 9;
    float x0 = fv[0], y0 = fv[1], z0 = fv[2];
    float x1 = fv[3], y1 = fv[4], z1 = fv[5];
    float x2 = fv[6], y2 = fv[7], z2 = fv[8];

    float w0 = yp * (x2 - x1) + xp * (y1 - y2) + (x1 * y2 - x2 * y1);
    float w1 = yp * (x0 - x2) + xp * (y2 - y0) + (x2 * y0 - x0 * y2);
    float w2 = yp * (x1 - x0) + xp * (y0 - y1) + (x0 * y1 - x1 * y0);
    float det = w0 + w1 + w2;
    float sd = (__builtin_fabsf(det) > TINYF) ? det : 1.0f;
    float n0 = w0 / sd, n1 = w1 / sd, n2 = w2 / sd;
    n0 = fminf(fmaxf(n0, 0.0f), 1.0f);
    n1 = fminf(fmaxf(n1, 0.0f), 1.0f);
    n2 = fminf(fmaxf(n2, 0.0f), 1.0f);
    float s = n0 + n1 + n2;
    s = (s > TINYF) ? s : 1.0f;
    n0 /= s; n1 /= s; n2 /= s;
    float inv = n0 / z0 + n1 / z1 + n2 / z2;
    inv = (__builtin_fabsf(inv) > TINYF) ? inv : 1.0f;
    float zp = 1.0f / inv;

    const float TM = (float)(TT - 1);
    float t0 = fminf(fmaxf(n0 * TM * zp / z0, 0.0f), TM - 0.001f);
    float t1 = fminf(fmaxf(n1 * TM * zp / z1, 0.0f), TM - 0.001f);
    float t2 = fminf(fmaxf(n2 * TM * zp / z2, 0.0f), TM - 0.001f);
    float l0 = floorf(t0), l1 = floorf(t1), l2 = floorf(t2);
    float f0 = t0 - l0, f1 = t1 - l1, f2 = t2 - l2;
    int i0 = (int)l0, i1 = (int)l1, i2 = (int)l2;

    const float* tb = tex + (size_t)(b * FF + f) * (TT * TT * TT * 3);
    float cr = 0.0f, cg = 0.0f, cb = 0.0f;
#pragma unroll
    for (int d0 = 0; d0 < 2; ++d0)
#pragma unroll
        for (int d1 = 0; d1 < 2; ++d1)
#pragma unroll
            for (int d2 = 0; d2 < 2; ++d2) {
                float wgt = (d0 ? f0 : 1.0f - f0) * (d1 ? f1 : 1.0f - f1) * (d2 ? f2 : 1.0f - f2);
                int idx = (i0 + d0) * (TT * TT) + (i1 + d1) * TT + (i2 + d2);
                const float* s3 = tb + idx * 3;
                cr += wgt * s3[0];
                cg += wgt * s3[1];
                cb += wgt * s3[2];
            }
    o[0] = cr; o[1] = cg; o[2] = cb; o[3] = 1.0f; o[4] = zp;
}

// ---------------------------------------------------------------------------
extern "C" void kernel_launch(void* const* d_in, const int* in_sizes, int n_in,
                              void* d_out, int out_size, void* d_ws, size_t ws_size,
                              hipStream_t stream) {
    (void)in_sizes; (void)n_in; (void)out_size; (void)ws_size;
    const float* faces = (const float*)d_in[0];  // [4,4096,3,3] f32
    const float* tex   = (const float*)d_in[1];  // [4,4096,4,4,4,3] f32
    float* out = (float*)d_out;                  // [4,256,256,5] f32

    float* prep = (float*)d_ws;                                   // 1 MiB
    int* fidx = (int*)((char*)d_ws + (size_t)BB * FF * 16 * 4);   // 1 MiB

    prep_kernel<<<(BB * FF) / 256, 256, 0, stream>>>(faces, prep);
    depth_kernel<<<BB * 128, 256, 0, stream>>>(prep, fidx);
    shade_kernel<<<(BB * IS * IS) / 256, 256, 0, stream>>>(faces, tex, fidx, out);
}